// BilinearPredictor_43009802502552
// MI455X (gfx1250) — compile-verified
//
#include <hip/hip_runtime.h>
#include <hip/hip_bf16.h>

typedef __attribute__((ext_vector_type(2))) float v2f;
typedef __attribute__((ext_vector_type(8))) float v8f;

#define HDIM  128
#define NCLS  4
#define GCOLS (NCLS * HDIM)   // 512

// ---------------------------------------------------------------------------
// Phase 1: g[v, c*128+f] = sum_k h[v,k] * W[c,f,k]
// One GEMM: A = h [n_nodes x 128], B = W viewed as [512 rows][128 k] (row-major,
// k innermost), contracted over k.  Per wave: 16x64 output tile via
// V_WMMA_F32_16X16X4_F32, 32 k-steps, 4 accumulators (A reused 4x).
// ---------------------------------------------------------------------------
__global__ __launch_bounds__(256)
void bilinear_gemm_wmma(const float* __restrict__ h,
                        const float* __restrict__ Wt,   // [512][128]
                        float* __restrict__ g,          // [n_nodes][512]
                        int n_nodes) {
  const int lane = threadIdx.x & 31;
  const int wave = threadIdx.x >> 5;   // 0..7 -> 64-col slab
  const int half = lane >> 4;          // K-half for A/B frags, M-half for D
  const int l15  = lane & 15;

  const int rowBase = blockIdx.x * 16;
  const int colBase = wave * 64;

  // A-fragment source row (ISA 16x4 f32 layout: lanes 0-15 M=0..15 K={0,1},
  // lanes 16-31 same M, K={2,3}).  Clamp keeps EXEC all-ones.
  int arow = rowBase + l15;
  if (arow >= n_nodes) arow = n_nodes - 1;
  const float* aptr = h + (size_t)arow * HDIM;

  const float* bptr0 = Wt + (size_t)(colBase +  0 + l15) * HDIM;
  const float* bptr1 = Wt + (size_t)(colBase + 16 + l15) * HDIM;
  const float* bptr2 = Wt + (size_t)(colBase + 32 + l15) * HDIM;
  const float* bptr3 = Wt + (size_t)(colBase + 48 + l15) * HDIM;

  v8f acc0 = {}, acc1 = {}, acc2 = {}, acc3 = {};

  #pragma unroll 4
  for (int k0 = 0; k0 < HDIM; k0 += 4) {
    const int k = k0 + half * 2;
    v2f a  = *(const v2f*)(aptr  + k);
    v2f b0 = *(const v2f*)(bptr0 + k);
    v2f b1 = *(const v2f*)(bptr1 + k);
    v2f b2 = *(const v2f*)(bptr2 + k);
    v2f b3 = *(const v2f*)(bptr3 + k);
    // 8-arg form: (neg_a, A, neg_b, B, c_mod, C, reuse_a, reuse_b)
    acc0 = __builtin_amdgcn_wmma_f32_16x16x4_f32(false, a, false, b0, (short)0, acc0, false, false);
    acc1 = __builtin_amdgcn_wmma_f32_16x16x4_f32(false, a, false, b1, (short)0, acc1, false, false);
    acc2 = __builtin_amdgcn_wmma_f32_16x16x4_f32(false, a, false, b2, (short)0, acc2, false, false);
    acc3 = __builtin_amdgcn_wmma_f32_16x16x4_f32(false, a, false, b3, (short)0, acc3, false, false);
  }

  // D layout (32-bit C/D 16x16): VGPR j -> M = j + 8*half, N = lane&15.
  #pragma unroll
  for (int j = 0; j < 8; ++j) {
    const int node = rowBase + j + half * 8;
    if (node < n_nodes) {
      float* grow = g + (size_t)node * GCOLS + colBase + l15;
      grow[ 0] = acc0[j];
      grow[16] = acc1[j];
      grow[32] = acc2[j];
      grow[48] = acc3[j];
    }
  }
}

// ---------------------------------------------------------------------------
// Phase 2: scores[e,c] = h[src[e]] . g[dst[e], c, :] + b[c]
// 4 edges per wave (8 lanes/edge), float4 streaming, 3-step butterfly reduce.
// Working set (h + g ~ 102 MB) is L2-resident on MI455X (192 MB L2).
// ---------------------------------------------------------------------------
__global__ __launch_bounds__(256)
void edge_score(const float* __restrict__ h,
                const float* __restrict__ g,
                const float* __restrict__ bias,
                const int* __restrict__ src,
                const int* __restrict__ dst,
                float* __restrict__ out,   // [n_edges][4]
                int n_edges) {
  const int lane  = threadIdx.x & 31;
  const int waveB = threadIdx.x >> 5;
  const long waveG = (long)blockIdx.x * 8 + waveB;
  const int  sub  = lane & 7;
  const long e    = waveG * 4 + (lane >> 3);
  if (e >= (long)n_edges) return;

  const long s = (long)src[e];
  const long d = (long)dst[e];

  const float4* hu = (const float4*)(h + s * (long)HDIM);    // 32 x float4
  const float4* gd = (const float4*)(g + d * (long)GCOLS);   // 128 x float4

  float a0 = 0.f, a1 = 0.f, a2 = 0.f, a3 = 0.f;
  #pragma unroll
  for (int i = 0; i < 4; ++i) {
    const int fi = sub * 4 + i;        // this lane covers floats [sub*16, sub*16+15]
    float4 x  = hu[fi];
    float4 y0 = gd[0 * 32 + fi];
    float4 y1 = gd[1 * 32 + fi];
    float4 y2 = gd[2 * 32 + fi];
    float4 y3 = gd[3 * 32 + fi];
    a0 += x.x * y0.x + x.y * y0.y + x.z * y0.z + x.w * y0.w;
    a1 += x.x * y1.x + x.y * y1.y + x.z * y1.z + x.w * y1.w;
    a2 += x.x * y2.x + x.y * y2.y + x.z * y2.z + x.w * y2.w;
    a3 += x.x * y3.x + x.y * y3.y + x.z * y3.z + x.w * y3.w;
  }

  // Butterfly reduce within each 8-lane group (masks 1,2,4 stay in-group).
  #pragma unroll
  for (int m = 1; m <= 4; m <<= 1) {
    a0 += __shfl_xor(a0, m);
    a1 += __shfl_xor(a1, m);
    a2 += __shfl_xor(a2, m);
    a3 += __shfl_xor(a3, m);
  }

  if (sub == 0) {
    float4 o;
    o.x = a0 + bias[0];
    o.y = a1 + bias[1];
    o.z = a2 + bias[2];
    o.w = a3 + bias[3];
    ((float4*)out)[e] = o;
  }
}

// ---------------------------------------------------------------------------
extern "C" void kernel_launch(void* const* d_in, const int* in_sizes, int n_in,
                              void* d_out, int out_size, void* d_ws, size_t ws_size,
                              hipStream_t stream) {
  const float* h   = (const float*)d_in[0];   // [N, 128]
  const float* W   = (const float*)d_in[1];   // [4, 128, 128] == [512][128]
  const float* b   = (const float*)d_in[2];   // [4]
  const int*   src = (const int*)d_in[3];     // [E]
  const int*   dst = (const int*)d_in[4];     // [E]
  float*       out = (float*)d_out;           // [E, 4]
  float*       g   = (float*)d_ws;            // [N, 512] scratch (~82 MB)

  const int n_nodes = in_sizes[0] / HDIM;
  const int n_edges = in_sizes[3];

  // Phase 1: WMMA GEMM -> g
  const int gridA = (n_nodes + 15) / 16;
  bilinear_gemm_wmma<<<gridA, 256, 0, stream>>>(h, W, g, n_nodes);

  // Phase 2: gather + dot (32 edges per 256-thread block)
  const int gridB = (n_edges + 31) / 32;
  edge_score<<<gridB, 256, 0, stream>>>(h, g, b, src, dst, out, n_edges);
}